// Controller_22952305230176
// MI455X (gfx1250) — compile-verified
//
#include <hip/hip_runtime.h>
#include <math.h>

#define T_STEPS 12
#define S_LEN   512
#define B_SZ    128
#define D_DIM   512
#define M_MOD   8

typedef float v2f __attribute__((ext_vector_type(2)));
typedef float v8f __attribute__((ext_vector_type(8)));

// ---------------------------------------------------------------------------
// Prep: transpose the two recurrent weight matrices so the per-step matvecs in
// the controller kernel read them coalesced (thread d reads W[di*D + d]).
//   WqcRT[di*D + do] = Wqc[do*(2D) + D + di]   (right half of Wqc, transposed)
//   W1T  [di*D + do] = W1 [do*D + di]
// ---------------------------------------------------------------------------
__global__ void prep_transpose(const float* __restrict__ Wqc,
                               const float* __restrict__ W1,
                               float* __restrict__ WqcRT,
                               float* __restrict__ W1T) {
  int idx = blockIdx.x * blockDim.x + threadIdx.x;
  const int total = D_DIM * D_DIM;
  if (idx < total) {
    int di = idx / D_DIM, dd = idx % D_DIM;
    WqcRT[idx] = Wqc[dd * (2 * D_DIM) + D_DIM + di];
  } else if (idx < 2 * total) {
    int k = idx - total;
    int di = k / D_DIM, dd = k % D_DIM;
    W1T[k] = W1[dd * D_DIM + di];
  }
}

// ---------------------------------------------------------------------------
// Batched f32 WMMA GEMM:  C[t][m,n] = sum_k A[t][m,k] * B[t][n,k] + bias[t][n]
// (i.e. A @ B^T + bias) using V_WMMA_F32_16X16X4_F32, one wave per 16x16 tile.
// A-frag (16x4): lanes 0-15 M=lane (K=k,k+1), lanes 16-31 M=lane-16 (K=k+2,k+3)
// B-frag (4x16): same split on K, N=lane&15.
// C/D (16x16):   VGPR v -> M = v + 8*(lane>=16), N = lane&15.
// ---------------------------------------------------------------------------
__global__ void wmma_gemm_abt(const float* __restrict__ A, int lda,
                              const float* __restrict__ Bm, int ldb,
                              const float* __restrict__ bias,
                              float* __restrict__ C, int ldc,
                              int mtiles, int ntiles, int K,
                              int strideA_t, int strideB_t,
                              int strideBias_t, int strideC_t) {
  int bx   = blockIdx.x;
  int t    = bx / (mtiles * ntiles);
  int rem  = bx % (mtiles * ntiles);
  int mT   = rem / ntiles;
  int nT   = rem % ntiles;
  const float* At    = A    + (size_t)t * strideA_t;
  const float* Bt    = Bm   + (size_t)t * strideB_t;
  const float* biast = bias + (size_t)t * strideBias_t;
  float*       Ct    = C    + (size_t)t * strideC_t;

  int lane = threadIdx.x & 31;
  int half = lane >> 4;     // 0: K offset 0..1, 1: K offset 2..3
  int lr   = lane & 15;
  int m    = mT * 16 + lr;
  int n    = nT * 16 + lr;

  v8f acc = {};
  for (int k = 0; k < K; k += 4) {
    v2f a = *(const v2f*)(At + (size_t)m * lda + k + 2 * half);
    v2f b = *(const v2f*)(Bt + (size_t)n * ldb + k + 2 * half);
    acc = __builtin_amdgcn_wmma_f32_16x16x4_f32(
        /*neg_a=*/false, a, /*neg_b=*/false, b,
        /*c_mod=*/(short)0, acc, /*reuse_a=*/false, /*reuse_b=*/false);
  }

  float bn   = biast[nT * 16 + lr];
  int   row0 = mT * 16 + half * 8;
  int   col  = nT * 16 + lr;
#pragma unroll
  for (int v = 0; v < 8; ++v) {
    Ct[(size_t)(row0 + v) * ldc + col] = acc[v] + bn;
  }
}

// ---------------------------------------------------------------------------
// Persistent controller: one workgroup per batch element b (recurrence is
// independent per b), 512 threads = 16 wave32s, all T steps with only
// workgroup barriers. lstm_seq (134 MB) is L2-resident (192 MB L2).
// ---------------------------------------------------------------------------
__global__ __launch_bounds__(512) void controller_kernel(
    const float* __restrict__ lstm,    // [S,B,D]
    const int*   __restrict__ seqlen,  // [B]
    const float* __restrict__ gumbel,  // [T,B,M]
    const float* __restrict__ Apre,    // [T,B,D] (ws): q-path + bqc, prefolded
    const float* __restrict__ WqcRT,   // [D,D]   (ws)
    const float* __restrict__ W1T,     // [D,D]   (ws)
    const float* __restrict__ b1,      // [D]
    const float* __restrict__ W2,      // [M,D]
    const float* __restrict__ b2,      // [M]
    const float* __restrict__ w_cv,    // [D]
    const float* __restrict__ b_cv,    // scalar
    const float* __restrict__ c_init,  // [D]
    float* __restrict__ out) {
  const int b    = blockIdx.x;
  const int tid  = threadIdx.x;
  const int lane = tid & 31;
  const int wave = tid >> 5;

  __shared__ float sC[D_DIM];     // carried context c_prev
  __shared__ float sCq[D_DIM];    // cq
  __shared__ float sCqw[D_DIM];   // cq * w_cv
  __shared__ float sH[D_DIM];     // relu hidden
  __shared__ float sRaw[S_LEN];   // raw logits over S, then cv
  __shared__ float sLogit[M_MOD];
  __shared__ float sRed[16];
  __shared__ float sBcast[2];
  __shared__ int   sArg;

  float* out_logits = out;
  float* out_probs  = out + (size_t)T_STEPS * B_SZ * M_MOD;
  float* out_cs     = out + (size_t)2 * T_STEPS * B_SZ * M_MOD;
  float* out_cvs    = out_cs + (size_t)T_STEPS * B_SZ * D_DIM;

  const int   L   = seqlen[b];
  const float bcv = b_cv[0];
  const float wcv = w_cv[tid];

  sC[tid] = c_init[tid];
  __syncthreads();

  for (int t = 0; t < T_STEPS; ++t) {
    // ---- phase 1: cq = Apre[t,b] + c_prev @ WqcR^T (coalesced weight reads)
    {
      float acc = Apre[((size_t)t * B_SZ + b) * D_DIM + tid];
#pragma unroll 8
      for (int di = 0; di < D_DIM; ++di)
        acc += sC[di] * WqcRT[(size_t)di * D_DIM + tid];
      sCq[tid]  = acc;
      sCqw[tid] = acc * wcv;
    }
    __syncthreads();

    // ---- phase 2: h = relu(cq @ W1^T + b1)
    {
      float acc = b1[tid];
#pragma unroll 8
      for (int di = 0; di < D_DIM; ++di)
        acc += sCq[di] * W1T[(size_t)di * D_DIM + tid];
      sH[tid] = acc > 0.f ? acc : 0.f;
    }
    __syncthreads();

    // ---- phase 3: logits (wave m computes logit[m], M=8 <= 16 waves)
    if (wave < M_MOD) {
      float p = 0.f;
#pragma unroll
      for (int j = 0; j < D_DIM / 32; ++j) {
        int d = j * 32 + lane;
        p += sH[d] * W2[wave * D_DIM + d];
      }
#pragma unroll
      for (int off = 16; off > 0; off >>= 1) p += __shfl_xor(p, off, 32);
      if (lane == 0) {
        float lg = p + b2[wave];
        out_logits[((size_t)t * B_SZ + b) * M_MOD + wave] = lg;
        sLogit[wave] = lg + gumbel[((size_t)t * B_SZ + b) * M_MOD + wave];
      }
    }
    __syncthreads();
    // straight-through forward value is exactly one_hot(argmax(logit+g))
    if (tid == 0) {
      int am = 0;
      float best = sLogit[0];
#pragma unroll
      for (int m = 1; m < M_MOD; ++m)
        if (sLogit[m] > best) { best = sLogit[m]; am = m; }
      sArg = am;
    }
    __syncthreads();
    if (tid < M_MOD)
      out_probs[((size_t)t * B_SZ + b) * M_MOD + tid] = (tid == sArg) ? 1.f : 0.f;

    // ---- phase 4: raw[s] = <lstm[s,b,:], cq*w_cv> + b_cv, length-masked
    {
      float xw[16];
#pragma unroll
      for (int j = 0; j < 16; ++j) xw[j] = sCqw[j * 32 + lane];
      for (int i = 0; i < 32; ++i) {
        int s = wave * 32 + i;
        const float* row = lstm + ((size_t)s * B_SZ + b) * D_DIM;
        float p = 0.f;
#pragma unroll
        for (int j = 0; j < 16; ++j) p += row[j * 32 + lane] * xw[j];
#pragma unroll
        for (int off = 16; off > 0; off >>= 1) p += __shfl_xor(p, off, 32);
        if (lane == 0) sRaw[s] = (s < L) ? (p + bcv) : -INFINITY;
      }
    }
    __syncthreads();

    // ---- phase 5: softmax over S (block reduce max, then sum)
    {
      float r  = sRaw[tid];
      float mx = r;
#pragma unroll
      for (int off = 16; off > 0; off >>= 1) mx = fmaxf(mx, __shfl_xor(mx, off, 32));
      if (lane == 0) sRed[wave] = mx;
      __syncthreads();
      if (tid < 16) {
        float v = sRed[tid];
#pragma unroll
        for (int off = 8; off > 0; off >>= 1) v = fmaxf(v, __shfl_xor(v, off, 16));
        if (tid == 0) sBcast[0] = v;
      }
      __syncthreads();
      mx = sBcast[0];
      float e  = expf(r - mx);  // masked rows: exp(-inf)=0
      float sm = e;
#pragma unroll
      for (int off = 16; off > 0; off >>= 1) sm += __shfl_xor(sm, off, 32);
      if (lane == 0) sRed[wave] = sm;
      __syncthreads();
      if (tid < 16) {
        float v = sRed[tid];
#pragma unroll
        for (int off = 8; off > 0; off >>= 1) v += __shfl_xor(v, off, 16);
        if (tid == 0) sBcast[1] = v;
      }
      __syncthreads();
      float cv = e / sBcast[1];
      sRaw[tid] = cv;  // reuse as cv
      out_cvs[((size_t)t * B_SZ + b) * S_LEN + tid] = cv;
    }
    __syncthreads();

    // ---- phase 6: c[d] = sum_s lstm[s,b,d] * cv[s]  (coalesced over d)
    {
      float acc = 0.f;
#pragma unroll 4
      for (int s = 0; s < S_LEN; ++s)
        acc += lstm[((size_t)s * B_SZ + b) * D_DIM + tid] * sRaw[s];
      sC[tid] = acc;
      out_cs[((size_t)t * B_SZ + b) * D_DIM + tid] = acc;
    }
    __syncthreads();
  }
}

// ---------------------------------------------------------------------------
extern "C" void kernel_launch(void* const* d_in, const int* in_sizes, int n_in,
                              void* d_out, int out_size, void* d_ws, size_t ws_size,
                              hipStream_t stream) {
  (void)in_sizes; (void)n_in; (void)out_size;

  const float* lstm   = (const float*)d_in[0];   // [S,B,D]
  const float* qenc   = (const float*)d_in[1];   // [B,D]
  /* d_in[2] embed_seq unused by reference */
  const int*   seqlen = (const int*)d_in[3];     // [B]
  const float* gumbel = (const float*)d_in[4];   // [T,B,M]
  const float* Wq     = (const float*)d_in[5];   // [T,D,D]
  const float* bq     = (const float*)d_in[6];   // [T,D]
  const float* Wqc    = (const float*)d_in[7];   // [D,2D]
  const float* bqc    = (const float*)d_in[8];   // [D]
  const float* W1     = (const float*)d_in[9];   // [D,D]
  const float* b1     = (const float*)d_in[10];  // [D]
  const float* W2     = (const float*)d_in[11];  // [M,D]
  const float* b2     = (const float*)d_in[12];  // [M]
  const float* w_cv   = (const float*)d_in[13];  // [D]
  const float* b_cv   = (const float*)d_in[14];  // scalar
  const float* c_init = (const float*)d_in[15];  // [D]
  float* out = (float*)d_out;

  // workspace layout (8 MB total, floats)
  float* ws    = (float*)d_ws;
  float* Q     = ws;                                    // T*B*D
  float* Apre  = Q     + (size_t)T_STEPS * B_SZ * D_DIM; // T*B*D
  float* WqcRT = Apre  + (size_t)T_STEPS * B_SZ * D_DIM; // D*D
  float* W1T   = WqcRT + (size_t)D_DIM * D_DIM;          // D*D
  (void)ws_size;  // requires >= 8 MiB

  // 1) transpose recurrent weights for coalesced matvec reads
  prep_transpose<<<(2 * D_DIM * D_DIM + 255) / 256, 256, 0, stream>>>(
      Wqc, W1, WqcRT, W1T);

  // 2) Q[t] = q_encoding @ Wq[t]^T + bq[t]        (f32 WMMA)
  const int mt = B_SZ / 16, nt = D_DIM / 16;
  wmma_gemm_abt<<<T_STEPS * mt * nt, 32, 0, stream>>>(
      qenc, D_DIM, Wq, D_DIM, bq, Q, D_DIM, mt, nt, D_DIM,
      /*sA*/ 0, /*sB*/ D_DIM * D_DIM, /*sBias*/ D_DIM, /*sC*/ B_SZ * D_DIM);

  // 3) Apre[t] = Q[t] @ WqcL^T + bqc              (f32 WMMA; bqc prefolded)
  wmma_gemm_abt<<<T_STEPS * mt * nt, 32, 0, stream>>>(
      Q, D_DIM, Wqc, 2 * D_DIM, bqc, Apre, D_DIM, mt, nt, D_DIM,
      /*sA*/ B_SZ * D_DIM, /*sB*/ 0, /*sBias*/ 0, /*sC*/ B_SZ * D_DIM);

  // 4) the sequential controller, one workgroup per batch element
  controller_kernel<<<B_SZ, D_DIM, 0, stream>>>(
      lstm, seqlen, gumbel, Apre, WqcRT, W1T, b1, W2, b2, w_cv, b_cv, c_init,
      out);
}